// ColorLoss_5712306504108
// MI455X (gfx1250) — compile-verified
//
#include <hip/hip_runtime.h>
#include <cmath>

// ColorLoss on MI455X (gfx1250):
//   out = mean(|blur((x+1)/2) - blur((y+1)/2)|) = mean(|blur((x-y)/2)|)
// Separable 45-tap Gaussian as banded-matrix WMMA:
//   16x16 tile = Band(16x60) x Data(60x16), 15 x v_wmma_f32_16x16x4_f32.
// Two tiles per wave (shared band operand), bulk LDS operand loads,
// straight accumulating WMMA chains. Pass-2 interior staging uses
// global_load_async_to_lds_b128 (ASYNCcnt). t (100 MB) stays L2-resident.

#define KS      45
#define PAD     22
#define H       512
#define W       512
#define PLANES  96                    // 32 batch * 3 channels
#define NTOT    (PLANES * H * W)      // 25,165,824 elements
#define NCHUNK  15                    // 60-wide K window / 4
#define BT      80                    // band table copy stride (2 copies)

typedef float v2f __attribute__((ext_vector_type(2)));
typedef float v8f __attribute__((ext_vector_type(8)));

struct GaussW { float g[KS]; };

__device__ __forceinline__ int reflect512(int i) {
    i = (i < 0) ? -i : i;            // jnp.pad 'reflect': -1 -> 1
    i = (i > 511) ? 1022 - i : i;    // 512 -> 510
    return i;
}

// Zero-padded band table: bt[copy*BT + 15 + j] = g[j], zero outside [0,45).
// Two copies so the two 16-lane halves read disjoint LDS bank windows.
__device__ __forceinline__ void fill_band(float* bt, const GaussW& gw, int tid) {
    for (int i = tid; i < 2 * BT; i += 256) {
        int j = (i >= BT ? i - BT : i) - 15;
        bt[i] = ((unsigned)j < (unsigned)KS) ? gw.g[j] : 0.0f;
    }
}

// ---------------- Pass 1: vertical blur of d=(x-y)/2, 32h x 128w per WG ---
#define K1_STRIDE 136   // 128+8 pad: half1 row offset 2*136 == +16 mod 64 banks
__global__ __launch_bounds__(256)
void k_vblur(const float* __restrict__ x, const float* __restrict__ y,
             float* __restrict__ t, GaussW gw) {
    __shared__ __align__(16) float lds[76 * K1_STRIDE];
    __shared__ float bt[2 * BT];
    const int tid   = threadIdx.x;
    const int plane = blockIdx.z;
    const int h0    = blockIdx.y * 32;
    const int w0    = blockIdx.x * 128;
    const size_t pbase = (size_t)plane * (H * W);

    fill_band(bt, gw, tid);
    // Stage rows h0-22..h0+53 (reflected), 128 cols; fused diff; float4 path.
    const float4* x4 = (const float4*)(x + pbase + w0);
    const float4* y4 = (const float4*)(y + pbase + w0);
    float4* l4 = (float4*)lds;
    for (int idx = tid; idx < 76 * 32; idx += 256) {
        int j = idx >> 5;
        int q = idx & 31;
        size_t ga = (size_t)reflect512(h0 - PAD + j) * (W / 4) + q;
        float4 a = x4[ga], b = y4[ga], d;
        d.x = (a.x - b.x) * 0.5f;  d.y = (a.y - b.y) * 0.5f;
        d.z = (a.z - b.z) * 0.5f;  d.w = (a.w - b.w) * 0.5f;
        l4[j * (K1_STRIDE / 4) + q] = d;
    }
    __syncthreads();

    const int wave = tid >> 5;
    const int lane = tid & 31;
    const int half = lane >> 4;      // lanes 16-31 hold K+2 / M+8 halves
    const int mn   = lane & 15;      // M for A(band), N for B(data)/D
    const int wloc = wave * 16;

    // A = banded Gaussian A[m,j]=g[j-m]: load once, reuse for both tiles.
    const float* aP = &bt[BT * half + 15 + 2 * half - mn];
    v2f av[NCHUNK];
#pragma unroll
    for (int c = 0; c < NCHUNK; ++c) {
        av[c].x = aP[4 * c];
        av[c].y = aP[4 * c + 1];
    }
    float* tp = t + pbase;
    const int wcol = w0 + wloc + mn;
#pragma unroll
    for (int tt = 0; tt < 2; ++tt) {   // two 16-row tiles per wave
        const float* bP = &lds[(16 * tt + 2 * half) * K1_STRIDE + wloc + mn];
        v2f bv[NCHUNK];
#pragma unroll
        for (int c = 0; c < NCHUNK; ++c) {
            bv[c].x = bP[(4 * c)     * K1_STRIDE];
            bv[c].y = bP[(4 * c + 1) * K1_STRIDE];
        }
        v8f acc = {};
#pragma unroll
        for (int c = 0; c < NCHUNK; ++c)
            acc = __builtin_amdgcn_wmma_f32_16x16x4_f32(false, av[c], false, bv[c],
                                                        (short)0, acc, false, false);
        // D layout: vgpr r -> row M = r + 8*half, col N = mn
#pragma unroll
        for (int r = 0; r < 8; ++r)
            tp[(size_t)(h0 + 16 * tt + r + 8 * half) * W + wcol] = acc[r];
    }
}

// ---------------- Pass 2: horizontal blur + |.| sum, 128h x 32w per WG ----
#define K2_STRIDE 84    // bank = (20m + col) mod 64 -> distinct; halves disjoint
__global__ __launch_bounds__(256)
void k_hblur_abssum(const float* __restrict__ t, float* __restrict__ partials,
                    GaussW gw) {
    __shared__ __align__(16) float lds[128 * K2_STRIDE];
    __shared__ float bt[2 * BT];
    __shared__ float wsum[8];
    const int tid   = threadIdx.x;
    const int plane = blockIdx.z;
    const int h0    = blockIdx.y * 128;
    const int w0    = blockIdx.x * 32;
    const size_t pbase = (size_t)plane * (H * W);

    fill_band(bt, gw, tid);
    // Stage 128 rows x 80 cols starting at wl = w0-24 (8-float aligned).
    // Compute uses staged cols 2..78 (== w0-22 .. w0+54).
    const int wl = w0 - 24;
    if (wl >= 0 && wl + 79 <= 511) {
        // Interior: async copy global -> LDS (ASYNCcnt), b128 per op.
        const float* tbase = t + pbase + wl;
        for (int idx = tid; idx < 128 * 20; idx += 256) {
            int r = idx / 20;
            int q = idx - r * 20;
            unsigned la = (unsigned)(size_t)&lds[r * K2_STRIDE + 4 * q];
            int voff = (int)(((h0 + r) * W + 4 * q) * sizeof(float));
            asm volatile("global_load_async_to_lds_b128 %0, %1, %2"
                         :: "v"(la), "v"(voff), "s"(tbase) : "memory");
        }
        asm volatile("s_wait_asynccnt 0x0" ::: "memory");
    } else {                            // edge blocks: scalar + reflect
        for (int idx = tid; idx < 128 * 80; idx += 256) {
            int r = idx / 80;
            int c = idx - r * 80;
            lds[r * K2_STRIDE + c] =
                t[pbase + (size_t)(h0 + r) * W + reflect512(wl + c)];
        }
    }
    __syncthreads();

    const int wave = tid >> 5;
    const int lane = tid & 31;
    const int half = lane >> 4;
    const int mn   = lane & 15;

    // B = banded B[j,n]=g[j-n]: load once, reuse for both tiles.
    const float* bP = &bt[BT * half + 15 + 2 * half - mn];
    v2f bv[NCHUNK];
#pragma unroll
    for (int c = 0; c < NCHUNK; ++c) {
        bv[c].x = bP[4 * c];
        bv[c].y = bP[4 * c + 1];
    }
    float s = 0.f;
#pragma unroll
    for (int tt = 0; tt < 2; ++tt) {   // two 16-col tiles per wave
        // A = data A[m,j] = t[h0+16*wave+m][w0+16*tt-22+j]
        const float* aP = &lds[(wave * 16 + mn) * K2_STRIDE + 2 + 16 * tt + 2 * half];
        v2f av[NCHUNK];
#pragma unroll
        for (int c = 0; c < NCHUNK; ++c) {
            av[c].x = aP[4 * c];
            av[c].y = aP[4 * c + 1];
        }
        v8f acc = {};
#pragma unroll
        for (int c = 0; c < NCHUNK; ++c)
            acc = __builtin_amdgcn_wmma_f32_16x16x4_f32(false, av[c], false, bv[c],
                                                        (short)0, acc, false, false);
#pragma unroll
        for (int r = 0; r < 8; ++r) s += fabsf(acc[r]);
    }
    // Deterministic fixed-order wave32 reduction.
    for (int off = 16; off > 0; off >>= 1) s += __shfl_xor(s, off, 32);
    if (lane == 0) wsum[wave] = s;
    __syncthreads();
    if (tid == 0) {
        float tot = 0.f;
        for (int wv = 0; wv < 8; ++wv) tot += wsum[wv];
        int wgid = (blockIdx.z * gridDim.y + blockIdx.y) * gridDim.x + blockIdx.x;
        partials[wgid] = tot;
    }
}

// ---------------- Final deterministic reduction --------------------------
__global__ __launch_bounds__(256)
void k_reduce(const float* __restrict__ partials, float* __restrict__ out, int n) {
    __shared__ float sm[256];
    float s = 0.f;
    for (int i = threadIdx.x; i < n; i += 256) s += partials[i];
    sm[threadIdx.x] = s;
    __syncthreads();
    for (int off = 128; off > 0; off >>= 1) {
        if ((int)threadIdx.x < off) sm[threadIdx.x] += sm[threadIdx.x + off];
        __syncthreads();
    }
    if (threadIdx.x == 0) out[0] = sm[0] * (1.0f / (float)NTOT);
}

extern "C" void kernel_launch(void* const* d_in, const int* in_sizes, int n_in,
                              void* d_out, int out_size, void* d_ws, size_t ws_size,
                              hipStream_t stream) {
    (void)in_sizes; (void)n_in; (void)out_size; (void)ws_size;
    const float* x = (const float*)d_in[0];
    const float* y = (const float*)d_in[1];
    float* t        = (float*)d_ws;  // 96*512*512 f32 = 100,663,296 B (L2-resident)
    float* partials = (float*)((char*)d_ws + (size_t)PLANES * H * W * sizeof(float));

    GaussW gw;
    {
        double m = (KS - 1) / 2.0, sum = 0.0, tmp[KS];
        for (int i = 0; i < KS; ++i) {
            tmp[i] = exp(-((i - m) * (i - m)) / (2.0 * 50.0 * 50.0));
            sum += tmp[i];
        }
        for (int i = 0; i < KS; ++i) gw.g[i] = (float)(tmp[i] / sum);
    }

    dim3 blk(256);
    dim3 g1(W / 128, H / 32, PLANES);   // 4 x 16 x 96 = 6144 WGs
    dim3 g2(W / 32, H / 128, PLANES);   // 16 x 4 x 96 = 6144 WGs
    k_vblur<<<g1, blk, 0, stream>>>(x, y, t, gw);
    k_hblur_abssum<<<g2, blk, 0, stream>>>(t, partials, gw);
    int nP = (int)(g2.x * g2.y * g2.z);
    k_reduce<<<dim3(1), blk, 0, stream>>>(partials, (float*)d_out, nP);
}